// CustomGCN_721554505840
// MI455X (gfx1250) — compile-verified
//
#include <hip/hip_runtime.h>
#include <math.h>

// Problem constants (from reference)
#define N_NODES 1024
#define IN_CH   1433
#define HID     16
#define OUT_CH  7
#define N_EDGES 4096
#define NE      (N_EDGES + N_NODES)   // 5120 edges incl. self-loops
#define TE_N    (NE - 1)              // 5119 embedded points

typedef __attribute__((ext_vector_type(2))) float v2f;
typedef __attribute__((ext_vector_type(8))) float v8f;

// ---------------------------------------------------------------------------
// Utility kernels
// ---------------------------------------------------------------------------
__global__ void zero_f32(float* __restrict__ p, int n) {
    int i = blockIdx.x * blockDim.x + threadIdx.x;
    if (i < n) p[i] = 0.0f;
}

__global__ void build_edges(const int* __restrict__ ei,
                            int* __restrict__ src, int* __restrict__ dst,
                            float* __restrict__ cnt) {
    int e = blockIdx.x * blockDim.x + threadIdx.x;
    if (e >= NE) return;
    int s, d;
    if (e < N_EDGES) { s = ei[e]; d = ei[N_EDGES + e]; }
    else             { s = e - N_EDGES; d = s; }
    src[e] = s;
    dst[e] = d;
    atomicAdd(&cnt[d], 1.0f);
}

// Zero-pad W2 (7x16) / b2 (7) into full 16x16 / 16 buffers so the WMMA GEMM
// inner loop is branch-free (no exec-mask toggles between WMMAs).
__global__ void pad_w2(const float* __restrict__ W2, const float* __restrict__ b2,
                       float* __restrict__ Wp, float* __restrict__ bp) {
    int i = threadIdx.x;              // 0..255, single block
    if (i < 16 * 16) {
        int r = i >> 4;
        Wp[i] = (r < OUT_CH) ? W2[r * HID + (i & 15)] : 0.0f;
    }
    if (i < 16) bp[i] = (i < OUT_CH) ? b2[i] : 0.0f;
}

// ---------------------------------------------------------------------------
// WMMA f32 GEMM: D[M x Cout] = A[M x K] @ W[16 x K]^T + bias  (rows >= Nstore
// of W must be zero-padded). One wave computes one 16x16 tile via
// V_WMMA_F32_16X16X4_F32; per K-step: one b64 load for A, one for B, one WMMA.
// ---------------------------------------------------------------------------
__global__ void gemm_wmma_f32(const float* __restrict__ A,
                              const float* __restrict__ W,
                              const float* __restrict__ bias,
                              float* __restrict__ D,
                              int K, int Nstore, int Cout) {
    const int mbase = blockIdx.x * 16;
    const int lane  = threadIdx.x;        // 0..31 (wave32)
    const int hi    = lane >> 4;          // 0 or 1
    const int lo    = lane & 15;
    const int koff  = 2 * hi;             // lane-half K offset within a step

    // loop-invariant row pointers (A fragment row = mbase+lo, B row = weight lo)
    const float* __restrict__ ap = A + (size_t)(mbase + lo) * K + koff;
    const float* __restrict__ bp = W + (size_t)lo * K + koff;

    v8f c = {};
    const int K4 = K & ~3;
    for (int k0 = 0; k0 < K4; k0 += 4) {
        v2f a, b;
        // A 16x4 f32 fragment: V0 = K{k0 | k0+2}, V1 = K{k0+1 | k0+3}
        a.x = ap[k0];
        a.y = ap[k0 + 1];
        // B 4x16 f32 fragment, same half-split: contiguous pair per lane
        b.x = bp[k0];
        b.y = bp[k0 + 1];
        c = __builtin_amdgcn_wmma_f32_16x16x4_f32(
                false, a, false, b, (short)0, c, false, false);
    }
    // K tail, applied directly on the C fragment:
    // element (lane, v) = C[mbase + v + 8*hi][lo]
    for (int k = K4; k < K; ++k) {
        float wv = W[(size_t)lo * K + k];
#pragma unroll
        for (int v = 0; v < 8; ++v) {
            int m = mbase + v + 8 * hi;
            c[v] += A[(size_t)m * K + k] * wv;
        }
    }
    if (lo < Nstore) {
        float bv = bias[lo];
#pragma unroll
        for (int v = 0; v < 8; ++v) {
            int m = mbase + v + 8 * hi;
            D[m * Cout + lo] = c[v] + bv;
        }
    }
}

// ---------------------------------------------------------------------------
// digamma for positive (integer-valued) float args via shift + asymptotic series
// ---------------------------------------------------------------------------
__device__ __forceinline__ float digammaf_dev(float x) {
    float r = 0.0f;
    while (x < 6.0f) { r -= 1.0f / x; x += 1.0f; }
    float xi  = 1.0f / x;
    float xi2 = xi * xi;
    return r + logf(x) - 0.5f * xi
           - xi2 * (1.0f / 12.0f - xi2 * (1.0f / 120.0f - xi2 * (1.0f / 252.0f)));
}

// ---------------------------------------------------------------------------
// KSG transfer entropy, k=1, Chebyshev metric.
// grid = (rowChunks, C). Each block stages the channel's gathered sequences in
// LDS (2 x 5120 floats = 40KB of the 320KB/WGP pool) and scans rows twice.
// sy[j]/sy[j+1] carried in a rolling register: 2 DS broadcasts per iteration.
// ---------------------------------------------------------------------------
__global__ void te_kernel(const float* __restrict__ h, int C,
                          const int* __restrict__ src, const int* __restrict__ dst,
                          float* __restrict__ tes_accum) {
    __shared__ float sx[NE];
    __shared__ float sy[NE];
    __shared__ float red[256];

    const int c   = blockIdx.y;
    const int tid = threadIdx.x;

    for (int e = tid; e < NE; e += blockDim.x) {
        sx[e] = h[dst[e] * C + c];   // x_t  = xi[:-1] via sx[0..n-1]
        sy[e] = h[src[e] * C + c];   // y_t  = sy[0..n-1], y_{t+1} = sy[1..n]
    }
    __syncthreads();

    float partial = 0.0f;
    for (int i = blockIdx.x * blockDim.x + tid; i < TE_N;
         i += gridDim.x * blockDim.x) {
        const float xi = sx[i], yi = sy[i], zi = sy[i + 1];

        // pass 1: eps_i = min_{j != i} max(|dx|,|dy|,|dz|)
        float eps = 3.402823466e38f;
        {
            float yn = sy[0];
#pragma unroll 4
            for (int j = 0; j < TE_N; ++j) {
                float yj = yn;
                yn = sy[j + 1];
                float ax = fabsf(xi - sx[j]);
                float ay = fabsf(yi - yj);
                float az = fabsf(zi - yn);
                float d  = fmaxf(fmaxf(ax, ay), az);
                eps = fminf(eps, (j == i) ? 3.402823466e38f : d);
            }
        }
        // pass 2: neighbor counts (self term included, matching the
        // reference's  sum(.) - 1  followed by  +1  before digamma)
        int cy = 0, cxy = 0, cyz = 0;
        {
            float yn = sy[0];
#pragma unroll 4
            for (int j = 0; j < TE_N; ++j) {
                float yj = yn;
                yn = sy[j + 1];
                float ax = fabsf(xi - sx[j]);
                float ay = fabsf(yi - yj);
                float az = fabsf(zi - yn);
                cy  += (ay < eps) ? 1 : 0;
                cxy += (fmaxf(ax, ay) < eps) ? 1 : 0;
                cyz += (fmaxf(ay, az) < eps) ? 1 : 0;
            }
        }
        partial += digammaf_dev((float)cy)
                 - digammaf_dev((float)cxy)
                 - digammaf_dev((float)cyz);
    }

    red[tid] = partial;
    __syncthreads();
    for (int s = blockDim.x >> 1; s > 0; s >>= 1) {
        if (tid < s) red[tid] += red[tid + s];
        __syncthreads();
    }
    if (tid == 0) atomicAdd(&tes_accum[c], red[0]);
}

// ---------------------------------------------------------------------------
// msg = sigmoid(te_c * h[src]); scatter-add into agg[dst]
// ---------------------------------------------------------------------------
__global__ void msg_agg(const float* __restrict__ h, int C,
                        const int* __restrict__ src, const int* __restrict__ dst,
                        const float* __restrict__ tes_accum, float scale,
                        float* __restrict__ agg) {
    int idx = blockIdx.x * blockDim.x + threadIdx.x;
    if (idx >= NE * C) return;
    int e = idx / C;
    int c = idx - e * C;
    // te = psi(1) + mean(...) , psi(1) = -EulerGamma
    float te = (-0.57721566490153286f + tes_accum[c] / (float)TE_N) * scale;
    float v  = te * h[src[e] * C + c];
    float m  = 1.0f / (1.0f + expf(-v));
    atomicAdd(&agg[dst[e] * C + c], m);
}

__global__ void finalize(float* __restrict__ agg, const float* __restrict__ cnt,
                         int C, int do_relu) {
    int idx = blockIdx.x * blockDim.x + threadIdx.x;
    if (idx >= N_NODES * C) return;
    int v  = idx / C;
    float x = agg[idx] / fmaxf(cnt[v], 1.0f);
    if (do_relu) x = fmaxf(x, 0.0f);
    agg[idx] = x;
}

__global__ void log_softmax_k(const float* __restrict__ h, float* __restrict__ out) {
    int v = blockIdx.x * blockDim.x + threadIdx.x;
    if (v >= N_NODES) return;
    float m = -3.402823466e38f;
#pragma unroll
    for (int c = 0; c < OUT_CH; ++c) m = fmaxf(m, h[v * OUT_CH + c]);
    float s = 0.0f;
#pragma unroll
    for (int c = 0; c < OUT_CH; ++c) s += expf(h[v * OUT_CH + c] - m);
    float l = logf(s);
#pragma unroll
    for (int c = 0; c < OUT_CH; ++c)
        out[v * OUT_CH + c] = h[v * OUT_CH + c] - m - l;
}

// ---------------------------------------------------------------------------
// Host launcher
// ---------------------------------------------------------------------------
extern "C" void kernel_launch(void* const* d_in, const int* in_sizes, int n_in,
                              void* d_out, int out_size, void* d_ws, size_t ws_size,
                              hipStream_t stream) {
    const float* x  = (const float*)d_in[0];
    const int*   ei = (const int*)  d_in[1];
    const float* W1 = (const float*)d_in[2];
    const float* b1 = (const float*)d_in[3];
    const float* W2 = (const float*)d_in[4];
    const float* b2 = (const float*)d_in[5];
    float* out = (float*)d_out;

    // workspace carve-up (256B aligned)
    char* w = (char*)d_ws;
    size_t off = 0;
    auto carve = [&](size_t bytes) {
        void* p = w + off;
        off += (bytes + 255) & ~(size_t)255;
        return p;
    };
    int*   src  = (int*)  carve(NE * sizeof(int));
    int*   dst  = (int*)  carve(NE * sizeof(int));
    float* cnt  = (float*)carve(N_NODES * sizeof(float));
    float* tes  = (float*)carve(32 * sizeof(float));
    float* Wp   = (float*)carve(16 * 16 * sizeof(float));
    float* bp   = (float*)carve(16 * sizeof(float));
    float* h1   = (float*)carve((size_t)N_NODES * HID * sizeof(float));
    float* agg1 = (float*)carve((size_t)N_NODES * HID * sizeof(float));
    float* h2   = (float*)carve((size_t)N_NODES * OUT_CH * sizeof(float));
    float* agg2 = (float*)carve((size_t)N_NODES * OUT_CH * sizeof(float));
    (void)ws_size; (void)in_sizes; (void)n_in; (void)out_size;

    const int T = 256;

    // edges + in-degree counts + padded layer-2 weights
    zero_f32<<<(N_NODES + T - 1) / T, T, 0, stream>>>(cnt, N_NODES);
    build_edges<<<(NE + T - 1) / T, T, 0, stream>>>(ei, src, dst, cnt);
    pad_w2<<<1, 256, 0, stream>>>(W2, b2, Wp, bp);

    // ---- layer 1 ----
    gemm_wmma_f32<<<N_NODES / 16, 32, 0, stream>>>(x, W1, b1, h1, IN_CH, HID, HID);

    zero_f32<<<1, 32, 0, stream>>>(tes, 32);
    te_kernel<<<dim3(20, HID), T, 0, stream>>>(h1, HID, src, dst, tes);

    zero_f32<<<(N_NODES * HID + T - 1) / T, T, 0, stream>>>(agg1, N_NODES * HID);
    msg_agg<<<(NE * HID + T - 1) / T, T, 0, stream>>>(h1, HID, src, dst, tes, 1.0f, agg1);
    finalize<<<(N_NODES * HID + T - 1) / T, T, 0, stream>>>(agg1, cnt, HID, 1);

    // ---- layer 2 ----
    gemm_wmma_f32<<<N_NODES / 16, 32, 0, stream>>>(agg1, Wp, bp, h2, HID, OUT_CH, OUT_CH);

    zero_f32<<<1, 32, 0, stream>>>(tes, 32);
    te_kernel<<<dim3(20, OUT_CH), T, 0, stream>>>(h2, OUT_CH, src, dst, tes);

    zero_f32<<<(N_NODES * OUT_CH + T - 1) / T, T, 0, stream>>>(agg2, N_NODES * OUT_CH);
    msg_agg<<<(NE * OUT_CH + T - 1) / T, T, 0, stream>>>(h2, OUT_CH, src, dst, tes, 0.5f, agg2);
    finalize<<<(N_NODES * OUT_CH + T - 1) / T, T, 0, stream>>>(agg2, cnt, OUT_CH, 0);

    log_softmax_k<<<(N_NODES + T - 1) / T, T, 0, stream>>>(agg2, out);
}